// NeuronClass_56427280335092
// MI455X (gfx1250) — compile-verified
//
#include <hip/hip_runtime.h>

// Problem constants from the reference (shapes fixed by setup_inputs):
//   B=32, T=1024, N=2048;  ALPHA = 1 - 0.05/10 = 0.995, V_TH = 2.0
#define ALPHA_C 0.995f
#define VTH_C   2.0f

typedef float v2f __attribute__((ext_vector_type(2)));
typedef float v8f __attribute__((ext_vector_type(8)));

// ---------------------------------------------------------------------------
// Pass 1: S[row] = dot(x[row, 0:N], w) for row in [0, B*T)  (row = b*T + t)
// One wave per 16-row tile using V_WMMA_F32_16X16X4_F32.
//   A (16x4 f32):  lane<16 -> M=lane,   {K=0,K=1}; lane>=16 -> M=lane-16, {K=2,K=3}
//   B (4x16 f32):  w replicated across all 16 columns, so each lane slot only
//                  needs w[K]; slot K = k + 2*(lane>=16) + vgpr.
//   C (16x16 f32): vgpr g holds (M=g, N=lane) for lanes 0-15 and (M=8+g) for
//                  lanes 16-31 -> lane 0 carries rows 0..7, lane 16 rows 8..15.
// ---------------------------------------------------------------------------
__global__ __launch_bounds__(256) void lif_gemv_wmma(
    const float* __restrict__ x, const float* __restrict__ w,
    float* __restrict__ S, int rows, int N) {
  const int lane = threadIdx.x & 31;
  const int wave = threadIdx.x >> 5;
  const int tile = blockIdx.x * 8 + wave;     // 8 waves / block
  const int row0 = tile * 16;
  if (row0 >= rows) return;                    // wave-uniform: EXEC stays all-1s

  const int half = lane >> 4;                  // 0 or 1
  const int m    = lane & 15;                  // row within the 16-row tile
  const float* xrow = x + (long long)(row0 + m) * (long long)N;

  v8f c0 = {}; v8f c1 = {};
  for (int k = 0; k < N; k += 8) {
    const int ka = k + 2 * half;
    v2f a0 = *(const v2f*)(xrow + ka);
    v2f b0 = *(const v2f*)(w    + ka);
    v2f a1 = *(const v2f*)(xrow + ka + 4);
    v2f b1 = *(const v2f*)(w    + ka + 4);
    c0 = __builtin_amdgcn_wmma_f32_16x16x4_f32(false, a0, false, b0,
                                               (short)0, c0, false, false);
    c1 = __builtin_amdgcn_wmma_f32_16x16x4_f32(false, a1, false, b1,
                                               (short)0, c1, false, false);
  }
  v8f c = c0 + c1;
  if ((lane & 15) == 0) {                      // lanes 0 and 16
    float* dst = S + row0 + 8 * half;
#pragma unroll
    for (int i = 0; i < 8; ++i) dst[i] = c[i];
  }
}

// ---------------------------------------------------------------------------
// Pass 2: sequential LIF recurrence per batch (32 chains, one wave).
//   v = alpha*v + S[b,t] - vth*z ; z = (v > vth) ; U[b,t] = S[b,t] - v*(w.w)
// Writes v_seq and z_seq directly into d_out.
// ---------------------------------------------------------------------------
__global__ void lif_recurrence(
    const float* __restrict__ S, const float* __restrict__ w,
    float* __restrict__ vout, float* __restrict__ zout,
    float* __restrict__ U, int B, int T, int N) {
  __shared__ float red[32];
  const int lane = threadIdx.x;                // 32 threads
  // ww = dot(w, w) via LDS reduction
  float acc = 0.f;
  for (int i = lane; i < N; i += 32) { float wv = w[i]; acc = fmaf(wv, wv, acc); }
  red[lane] = acc;
  __syncthreads();
  if (lane == 0) {
    float t = 0.f;
    for (int i = 0; i < 32; ++i) t += red[i];
    red[0] = t;
  }
  __syncthreads();
  const float ww = red[0];

  if (lane >= B) return;
  const long long base = (long long)lane * T;
  float v = 0.f, z = 0.f;
  for (int t = 0; t < T; ++t) {
    const float st = S[base + t];
    v = fmaf(ALPHA_C, v, st) - VTH_C * z;      // z from previous step
    z = (v > VTH_C) ? 1.f : 0.f;
    vout[base + t] = v;
    zout[base + t] = z;
    U[base + t] = fmaf(-v, ww, st);            // eps . w
  }
}

// ---------------------------------------------------------------------------
// Pass 3: per-(b,n) independent trace/grad recurrence, streaming x.
//   eps = x - v*w_n ; g += v*eps + u*p (p pre-update) ; p = alpha*p + x
// Block = one batch b, 256 consecutive n -> coalesced; V/U loads are uniform
// per block (scalarized by the compiler). Prefetch hides the stride-N walk.
// ---------------------------------------------------------------------------
__global__ __launch_bounds__(256) void lif_grad(
    const float* __restrict__ x, const float* __restrict__ w,
    const float* __restrict__ V, const float* __restrict__ U,
    float* __restrict__ G, int B, int T, int N) {
  const int blocksPerB = N >> 8;               // 256 n per block
  const int b = blockIdx.x / blocksPerB;
  const int n = (blockIdx.x % blocksPerB) * 256 + threadIdx.x;
  const float wn = w[n];
  const long long xbase = ((long long)b * T) * (long long)N + n;
  const long long sbase = (long long)b * T;

  float p = 0.f, g = 0.f;
  for (int t = 0; t < T; ++t) {
    const float xv = x[xbase + (long long)t * N];
    if (t + 8 < T) __builtin_prefetch(&x[xbase + (long long)(t + 8) * N], 0, 0);
    const float vv = V[sbase + t];
    const float uu = U[sbase + t];
    const float eps = fmaf(-vv, wn, xv);
    g = fmaf(vv, eps, fmaf(uu, p, g));         // uses pre-update p
    p = fmaf(ALPHA_C, p, xv);
  }
  G[(long long)b * N + n] = g;
}

// ---------------------------------------------------------------------------
extern "C" void kernel_launch(void* const* d_in, const int* in_sizes, int n_in,
                              void* d_out, int out_size, void* d_ws, size_t ws_size,
                              hipStream_t stream) {
  const float* x = (const float*)d_in[0];      // [B, T, N] f32
  const float* w = (const float*)d_in[1];      // [N] f32

  const int N  = in_sizes[1];                  // 2048
  const long long BT = (long long)in_sizes[0] / N;  // 32768
  const int B = 32;
  const int T = (int)(BT / B);                 // 1024

  float* vout = (float*)d_out;                 // [B,T]
  float* zout = vout + BT;                     // [B,T]
  float* G    = vout + 2 * BT;                 // [B,N]

  float* S = (float*)d_ws;                     // [B*T] dot products
  float* U = S + BT;                           // [B*T] eps.w scalars

  const int tiles  = (int)(BT / 16);           // 2048 wave-tiles
  const int blocks = tiles / 8;                // 8 waves per block

  lif_gemv_wmma<<<blocks, 256, 0, stream>>>(x, w, S, (int)BT, N);
  lif_recurrence<<<1, 32, 0, stream>>>(S, w, vout, zout, U, B, T, N);
  lif_grad<<<B * (N >> 8), 256, 0, stream>>>(x, w, vout, U, G, B, T, N);
}